// IPC_26697516712457
// MI455X (gfx1250) — compile-verified
//
#include <hip/hip_runtime.h>

// CDNA5 / gfx1250: wave32, fp32 WMMA 16x16x4.
typedef __attribute__((ext_vector_type(2))) float v2f;
typedef __attribute__((ext_vector_type(8))) float v8f;

#ifndef FEAT
#define FEAT 16
#endif

// Per-vertex MLP: X(18x128) -> [8x18] relu -> [8x8] relu -> [1x8] -> out(128)
// One wave per vertex. WMMA tiles: M=16 (rows 0..7 valid), N=16 (8 tiles over 128 pts),
// K=4 per wmma (layer1: 5 chunks covering K=20 incl. padding; layer2: 2 chunks, K=8).
__global__ __launch_bounds__(256) void IPC_vertex_mlp_wmma(
    const float* __restrict__ vf,   // (N,16,128)
    const float* __restrict__ py,   // (N,128,2)
    const float* __restrict__ vp,   // (N,233)
    float* __restrict__ out,        // (N,128)
    int nvert)
{
    const int lane = threadIdx.x & 31;
    const int wave = threadIdx.x >> 5;
    const int v    = blockIdx.x * (blockDim.x >> 5) + wave;
    if (v >= nvert) return;                 // wave-uniform guard (grid sized exactly)

    const float* xf  = vf + (size_t)v * (FEAT * 128);
    const float* pyv = py + (size_t)v * (128 * 2);
    const float* p   = vp + (size_t)v * 233;

    const int  row   = lane & 15;           // M-row (A) or N-col (B/D) within tile
    const int  khalf = (lane >> 4) << 1;    // 0 for lanes 0-15, 2 for lanes 16-31
    const bool lo    = (lane < 16);

    // ---- per-channel min over the 128 points (c_py = py_pred - min, exact: RO=4 = 2^2) ----
    float m0 = __builtin_inff(), m1 = __builtin_inff();
#pragma unroll
    for (int i = 0; i < 4; ++i) {
        const int pt = lane + i * 32;
        m0 = fminf(m0, pyv[pt * 2 + 0]);
        m1 = fminf(m1, pyv[pt * 2 + 1]);
    }
#pragma unroll
    for (int off = 16; off > 0; off >>= 1) {
        m0 = fminf(m0, __shfl_xor(m0, off, 32));
        m1 = fminf(m1, __shfl_xor(m1, off, 32));
    }

    // ---- A operands: W1 (8x18) -> 5 chunks of 16x4; W2 (8x8) -> 2 chunks ----
    // A layout (ISA 7.12.2, 32-bit A 16x4): lane m(<16): {K=k0,k0+1}; lane 16+m: {K=k0+2,k0+3}
    v2f A1[5];
#pragma unroll
    for (int kc = 0; kc < 5; ++kc) {
        const int k0 = kc * 4 + khalf;
        const bool ok0 = (row < 8) && (k0 < 18);
        const bool ok1 = (row < 8) && (k0 + 1 < 18);
        float a0 = p[ok0 ? (row * 18 + k0)     : 0];
        float a1 = p[ok1 ? (row * 18 + k0 + 1) : 0];
        A1[kc].x = ok0 ? a0 : 0.0f;
        A1[kc].y = ok1 ? a1 : 0.0f;
    }
    v2f A2[2];
#pragma unroll
    for (int kc = 0; kc < 2; ++kc) {
        const int k0 = kc * 4 + khalf;      // 0..7, always valid column of W2
        const bool okr = (row < 8);
        float a0 = p[okr ? (152 + row * 8 + k0)     : 0];
        float a1 = p[okr ? (152 + row * 8 + k0 + 1) : 0];
        A2[kc].x = okr ? a0 : 0.0f;
        A2[kc].y = okr ? a1 : 0.0f;
    }

    float b1r[8], b2r[8], w3r[8];
#pragma unroll
    for (int r = 0; r < 8; ++r) {
        b1r[r] = p[144 + r];                // bias of layer 1
        b2r[r] = p[216 + r];                // bias of layer 2
        w3r[r] = p[224 + r];                // W3 (1x8)
    }
    const float b3 = p[232];

    // ---- 8 N-tiles of 16 points each ----
#pragma unroll 1
    for (int nt = 0; nt < 8; ++nt) {
        const int col = nt * 16 + row;      // point index handled by this lane (as B col)

        // ----- layer 1: C1 = W1 * X_tile -----
        v8f c1 = {};
#pragma unroll
        for (int kc = 0; kc < 5; ++kc) {
            const int k0 = kc * 4 + khalf;
            float bb0, bb1;
            if (kc < 4) {
                bb0 = xf[k0 * 128 + col];           // feature rows 0..15
                bb1 = xf[(k0 + 1) * 128 + col];
            } else {
                // rows 16/17 = shifted py channels (lanes 0-15); rows 18/19 = zero pad
                float p0 = pyv[col * 2 + 0];
                float p1 = pyv[col * 2 + 1];
                bb0 = lo ? (p0 - m0) : 0.0f;
                bb1 = lo ? (p1 - m1) : 0.0f;
            }
            v2f B; B.x = bb0; B.y = bb1;
            c1 = __builtin_amdgcn_wmma_f32_16x16x4_f32(
                     false, A1[kc], false, B, (short)0, c1, false, false);
        }
        // bias + relu (rows 8..15 are padding garbage, never consumed)
#pragma unroll
        for (int r = 0; r < 8; ++r) c1[r] = fmaxf(c1[r] + b1r[r], 0.0f);

        // ----- layer 2: C2 = W2 * H1_tile -----
        // H1[k][16nt+n] lives in lane n (0..15), accumulator VGPR k. Re-stripe into
        // B layout with wave shuffles (ds_bpermute on gfx1250).
        v8f c2 = {};
#pragma unroll
        for (int kc = 0; kc < 2; ++kc) {
            float s0 = __shfl(c1[kc * 4 + 0], row, 32);
            float s1 = __shfl(c1[kc * 4 + 1], row, 32);
            float s2 = __shfl(c1[kc * 4 + 2], row, 32);
            float s3 = __shfl(c1[kc * 4 + 3], row, 32);
            v2f B; B.x = lo ? s0 : s2; B.y = lo ? s1 : s3;
            c2 = __builtin_amdgcn_wmma_f32_16x16x4_f32(
                     false, A2[kc], false, B, (short)0, c2, false, false);
        }
#pragma unroll
        for (int r = 0; r < 8; ++r) c2[r] = fmaxf(c2[r] + b2r[r], 0.0f);

        // ----- layer 3: out = w3 . h2 + b3 (lanes 0-15 hold the 16 columns) -----
        float acc = b3;
#pragma unroll
        for (int r = 0; r < 8; ++r) acc = fmaf(w3r[r], c2[r], acc);
        if (lo) out[(size_t)v * 128 + nt * 16 + row] = acc;
    }
}

extern "C" void kernel_launch(void* const* d_in, const int* in_sizes, int n_in,
                              void* d_out, int out_size, void* d_ws, size_t ws_size,
                              hipStream_t stream) {
    (void)n_in; (void)out_size; (void)d_ws; (void)ws_size;
    const float* vf = (const float*)d_in[0];   // vertex_feature (N,16,128)
    const float* py = (const float*)d_in[1];   // py_pred        (N,128,2)
    const float* vp = (const float*)d_in[2];   // vertex_params  (N,233,1)
    float* out = (float*)d_out;                // (N,128)

    const int nvert = in_sizes[0] / (FEAT * 128);   // 16384
    const int wavesPerBlock = 8;                    // 256 threads = 8 wave32
    dim3 block(32 * wavesPerBlock);
    dim3 grid((nvert + wavesPerBlock - 1) / wavesPerBlock);
    IPC_vertex_mlp_wmma<<<grid, block, 0, stream>>>(vf, py, vp, out, nvert);
}